// QuantumParity_4312147165315
// MI455X (gfx1250) — compile-verified
//
#include <hip/hip_runtime.h>
#include <hip/hip_bf16.h>

// QuantumParity closed form (CNOT chain => Gray-code bijection => product
// distribution over flipped bits):
//   feats[i]     = prod_{k=0..i}   cos(x_k)    (i = 0..7)
//   feats[(i,j)] = prod_{k=i+1..j} cos(x_k)    (i<j, 28 pairs)
//   out[m][n]    = sum_k feats[m][k] * W[n][k] + b[n]
//
// Roofline: ~4.2 MB x-in + 1 MB out @ 23.3 TB/s => ~0.22 us floor; compute is
// ~30 MFLOP total => pure bandwidth. The 36->2 contraction rides
// V_WMMA_F32_16X16X4_F32 at full fp32. Features are staged through LDS so the
// half-wave-dependent A-fragment K-offset folds into an address register
// (constant ds_load offsets, no private-array selects, no scratch).

#define NQ      8
#define NFEAT   36          // = 9 K-steps of 4, no padding needed
#define FSTRIDE 44          // LDS row stride in floats: 176B, 16B-aligned
#define WAVES   8           // 256-thread blocks
#define SPW     32          // samples per wave (1 per lane, 2 WMMA tiles)

typedef __attribute__((ext_vector_type(2))) float v2f;
typedef __attribute__((ext_vector_type(8))) float v8f;

#if defined(__gfx1250__) && __has_builtin(__builtin_amdgcn_wmma_f32_16x16x4_f32)
#define HAVE_WMMA_F32X4 1
#endif

__global__ __launch_bounds__(256) void qparity_wmma_kernel(
    const float* __restrict__ x,     // (B, 8)
    const float* __restrict__ W,     // (2, 36)
    const float* __restrict__ bvec,  // (2,)
    float* __restrict__ out,         // (B, 2)
    int nSamples)
{
  __shared__ float lds[WAVES * SPW * FSTRIDE];   // 45056 B of the 320 KB WGP pool

  const int lane = (int)(threadIdx.x & 31u);
  const int wave = (int)(threadIdx.x >> 5u);
  const int waveBase = ((int)blockIdx.x * WAVES + wave) * SPW;  // first sample
  const int m    = lane & 15;          // row within a 16x16 WMMA tile
  const int sel2 = (lane >> 4) << 1;   // 0 (lanes 0-15) / 2 (lanes 16-31): K offset
  const int n    = lane & 15;          // output column carried in B/D frags

  // ---------------- per-lane sample: load x, cosines, 36 features ----------
  const int sample = waveBase + lane;                 // fully coalesced
  const float4 xa = *(const float4*)(x + (size_t)sample * NQ);
  const float4 xb = *(const float4*)(x + (size_t)sample * NQ + 4);
  float c[NQ];
  c[0] = cosf(xa.x); c[1] = cosf(xa.y); c[2] = cosf(xa.z); c[3] = cosf(xa.w);
  c[4] = cosf(xb.x); c[5] = cosf(xb.y); c[6] = cosf(xb.z); c[7] = cosf(xb.w);

  float feats[NFEAT];                 // all constant indices -> VGPRs only
  {
    float p = 1.0f;
#pragma unroll
    for (int k = 0; k < NQ; ++k) { p *= c[k]; feats[k] = p; }       // <Z_k>
    int idx = NQ;
#pragma unroll
    for (int i = 0; i < NQ; ++i) {
      float r = 1.0f;
#pragma unroll
      for (int j = i + 1; j < NQ; ++j) { r *= c[j]; feats[idx++] = r; } // <Z_iZ_j>
    }
  }

  // ---------------- stage features to LDS (9 x ds_store_b128) --------------
  float* frow = &lds[(wave * SPW + lane) * FSTRIDE];
#pragma unroll
  for (int t = 0; t < NFEAT / 4; ++t)
    *(float4*)(frow + 4 * t) =
        make_float4(feats[4 * t], feats[4 * t + 1], feats[4 * t + 2], feats[4 * t + 3]);

  // ---------------- B fragments: 2 real columns of W, 14 zero columns ------
  // Base pointer absorbs the half-wave K offset -> constant-index loads only.
  float bx[9], by[9];
  if (n < 2) {
    const float* wp = W + n * NFEAT + sel2;           // 8B-aligned
#pragma unroll
    for (int t = 0; t < 9; ++t) {
      const float2 w2 = *(const float2*)(wp + 4 * t);
      bx[t] = w2.x; by[t] = w2.y;
    }
  } else {
#pragma unroll
    for (int t = 0; t < 9; ++t) { bx[t] = 0.0f; by[t] = 0.0f; }
  }
  const float bn = (n < 2) ? bvec[n] : 0.0f;

  __syncthreads();                                    // LDS tile visible

  // ---------------- two 16x16 D tiles, 9 K-steps each, interleaved ---------
  // A frag addresses: row (m + 16*tile), K base folded in via sel2.
  const float* arow0 = &lds[(wave * SPW + m)      * FSTRIDE + sel2];
  const float* arow1 = &lds[(wave * SPW + m + 16) * FSTRIDE + sel2];

  v8f acc0 = {};
  v8f acc1 = {};
#if HAVE_WMMA_F32X4
#pragma unroll
  for (int t = 0; t < 9; ++t) {
    const int k = 4 * t;
    const float2 a0 = *(const float2*)(arow0 + k);    // ds_load_b64, 8B-aligned
    const float2 a1 = *(const float2*)(arow1 + k);
    v2f af0; af0.x = a0.x; af0.y = a0.y;
    v2f af1; af1.x = a1.x; af1.y = a1.y;
    v2f bf;  bf.x  = bx[t]; bf.y = by[t];
    // 8 args: (neg_a, A, neg_b, B, c_mod, C, reuse_a, reuse_b)
    acc0 = __builtin_amdgcn_wmma_f32_16x16x4_f32(
        false, af0, false, bf, (short)0, acc0, false, false);
    acc1 = __builtin_amdgcn_wmma_f32_16x16x4_f32(
        false, af1, false, bf, (short)0, acc1, false, false);
  }
#else
  // Compile-insurance fallback (not the gfx1250 path).
#pragma unroll
  for (int t = 0; t < 9; ++t) {
    const int k = 4 * t;
    acc0[0] = fmaf(arow0[k], bx[t], acc0[0]);
    acc1[0] = fmaf(arow1[k], by[t], acc1[0]);
  }
#endif

  // ---------------- writeback: D VGPR r = row r (+8 for high half) ---------
  if (n < 2) {
    const int r0 = waveBase + (sel2 ? 8 : 0);
#pragma unroll
    for (int r = 0; r < 8; ++r) {
      out[(size_t)(r0 + r)      * 2 + n] = acc0[r] + bn;   // tile 0: rows 0-15
      out[(size_t)(r0 + r + 16) * 2 + n] = acc1[r] + bn;   // tile 1: rows 16-31
    }
  }
}

extern "C" void kernel_launch(void* const* d_in, const int* in_sizes, int n_in,
                              void* d_out, int out_size, void* d_ws, size_t ws_size,
                              hipStream_t stream) {
  (void)n_in; (void)out_size; (void)d_ws; (void)ws_size;
  const float* x = (const float*)d_in[0];   // (B, 8)
  const float* W = (const float*)d_in[1];   // (2, 36)
  const float* b = (const float*)d_in[2];   // (2,)
  float* out = (float*)d_out;               // (B, 2)

  const int nSamples = in_sizes[0] / NQ;                  // 131072
  const int samplesPerBlock = WAVES * SPW;                // 256
  const int blocks = (nSamples + samplesPerBlock - 1) / samplesPerBlock;  // 512
  qparity_wmma_kernel<<<blocks, 256, 0, stream>>>(x, W, b, out, nSamples);
}